// IPABlock_73504070303814
// MI455X (gfx1250) — compile-verified
//
#include <hip/hip_runtime.h>
#include <hip/hip_bf16.h>
#include <math.h>

// ---------------------------------------------------------------------------
// IPA block for MI455X (gfx1250, wave32). All heavy math goes through
// V_WMMA_F32_16X16X4_F32 (fp32 matrix core path). Attention logits and the
// attention-apply step are expressed as batched GEMMs with concatenated K /
// N dimensions so that the pair bias, point attention, mask and pair-output
// contraction ride inside the matrix ops.
//
// GEMM v2: 256-thread blocks (8 waves), 128x128 block tile, 32x64 wave tile.
// LDS tiles stored as K-pairs (float2 per element) so each WMMA fragment is
// one conflict-free ds_load_b64 straight into an even VGPR pair (no moves).
// global_prefetch_b8 issued for the next K-block while computing.
// ---------------------------------------------------------------------------

typedef float v2f __attribute__((ext_vector_type(2)));
typedef float v8f __attribute__((ext_vector_type(8)));

#define B_  2
#define N_  1024
#define CS_ 384
#define CZ_ 128
#define C_  128
#define H_  8
#define PQ_ 8
#define PV_ 12
#define KC_ 409   // 128 (qk) + 24 (pts) + 256 (z bias) + 1 (mask)
#define VC_ 420   // 128 (v) + 36 (v_pts) + 256 (z2)

#define C1_       0.05103103630798288f   // sqrt(1/(3*128))
#define C2_       0.5773502691896258f    // sqrt(1/3)
#define HW_SCALE_ 0.09622504486493763f   // sqrt(1/(3*36))
#define MAX_ANG_  0.2617993877991494f    // 15 deg
#define MAX_TR_   1.5f

// ---- workspace layout (floats) --------------------------------------------
#define OFF_ROT  ((size_t)0)                      // B*N*9      = 18432
#define OFF_Q    (OFF_ROT  + 18432)               // B*N*H*C    = 2097152
#define OFF_KV   (OFF_Q    + 2097152)             // B*N*H*2C   = 4194304
#define OFF_QP   (OFF_KV   + 4194304)             // B*N*H*PQ*3 = 393216
#define OFF_KVP  (OFF_QP   + 393216)              // B*N*H*20*3 = 983040
#define OFF_ROWA (OFF_KVP  + 983040)              // B*H*N      = 16384
#define OFF_COLA (OFF_ROWA + 16384)               // B*H*N      = 16384
#define OFF_QC   (OFF_COLA + 16384)               // B*H*N*409  = 6701056
#define OFF_KC   (OFF_QC   + 6701056)             // 6701056
#define OFF_A    (OFF_KC   + 6701056)             // B*H*N*N    = 16777216
#define OFF_VC   (OFF_A    + 16777216)            // B*H*420*N  = 6881280
#define OFF_OC   (OFF_VC   + 6881280)             // B*H*N*420  = 6881280
#define OFF_F    (OFF_OC   + 6881280)             // B*N*2432   = 4980736
#define OFF_LN   (OFF_F    + 4980736)             // B*N*384    = 786432
#define OFF_BH   (OFF_LN   + 786432)              // B*N*128    = 262144
#define OFF_UPD  (OFF_BH   + 262144)              // B*N*6      = 12288
#define OFF_FH   (OFF_UPD  + 12288)               // B*N*1536   = 3145728
// total = 60,848,128 floats  (~243 MB of workspace)

__device__ __forceinline__ float softplusf(float x) {
  return (x > 20.f) ? x : log1pf(expf(x));
}

// ---------------------------------------------------------------------------
// Generic fp32 WMMA GEMM:  C[b] = A[b] (MxK) * op(B[b]) (KxN) + epilogue
// bNT=1 : B operand stored as (N x K) row-major (K contiguous)  ("NT")
// bNT=0 : B operand stored as (K x N) row-major (N contiguous)  ("NN")
// epilogue: + colAdd[n] + rowAdd[m] + resid[m,n]; act=1 -> exact GELU.
// ---------------------------------------------------------------------------
#define KRS 260   // LDS row stride in floats per k2-row (2*128 + 4 pad, even)

__global__ __launch_bounds__(256)
void wmma_gemm_f32(const float* __restrict__ A, long long Abs, int lda,
                   const float* __restrict__ B, long long Bbs, int ldb, int bNT,
                   float* __restrict__ C, long long Cbs, int ldc,
                   int M, int N, int K,
                   const float* __restrict__ colAdd, long long colBs,
                   const float* __restrict__ rowAdd, long long rowBs,
                   const float* __restrict__ resid, long long resBs, int ldres,
                   int act)
{
  // K-pair layout: element (k2, m) lives at [k2*KRS + 2*m (+ k&1)]
  __shared__ __align__(16) float As[8 * KRS];
  __shared__ __align__(16) float Bs[8 * KRS];

  const int bz = blockIdx.z;
  const float* Ab = A + (long long)bz * Abs;
  const float* Bb = B + (long long)bz * Bbs;
  float*       Cb = C + (long long)bz * Cbs;

  const int m0   = blockIdx.y * 128;
  const int n0   = blockIdx.x * 128;
  const int t    = threadIdx.x;       // 0..255
  const int wave = t >> 5;            // 0..7
  const int lane = t & 31;
  const int wm   = wave >> 1;         // 0..3 -> M offset wm*32
  const int wn   = wave & 1;          // 0..1 -> N offset wn*64
  const int lr   = lane & 15;
  const int kh   = (lane >> 4) << 1;  // 0 or 2 (ISA K-half select)

  v8f acc[2][4];
  for (int i = 0; i < 2; ++i)
    for (int s = 0; s < 4; ++s)
      for (int e = 0; e < 8; ++e) acc[i][s][e] = 0.0f;

  for (int k0 = 0; k0 < K; k0 += 16) {
    // ---- stage A: 128(m) x 16(k), 2048 elems, 8 per thread ----
    for (int i = 0; i < 8; ++i) {
      int idx = t + i * 256;
      int kk = idx & 15, mm = idx >> 4;
      int gm = m0 + mm, gk = k0 + kk;
      float va = (gm < M && gk < K) ? Ab[(long long)gm * lda + gk] : 0.0f;
      As[(kk >> 1) * KRS + 2 * mm + (kk & 1)] = va;
    }
    // ---- stage B ----
    if (bNT) {
      for (int i = 0; i < 8; ++i) {
        int idx = t + i * 256;
        int kk = idx & 15, nn = idx >> 4;
        int gn = n0 + nn, gk = k0 + kk;
        float vb = (gn < N && gk < K) ? Bb[(long long)gn * ldb + gk] : 0.0f;
        Bs[(kk >> 1) * KRS + 2 * nn + (kk & 1)] = vb;
      }
    } else {
      for (int i = 0; i < 8; ++i) {
        int idx = t + i * 256;
        int nn = idx & 127, kk = idx >> 7;
        int gn = n0 + nn, gk = k0 + kk;
        float vb = (gn < N && gk < K) ? Bb[(long long)gk * ldb + gn] : 0.0f;
        Bs[(kk >> 1) * KRS + 2 * nn + (kk & 1)] = vb;
      }
    }
    // ---- prefetch next K-block (global_prefetch_b8, speculative) ----
    if (k0 + 16 < K) {
      if (t < 128) {
        int gm = m0 + t;
        if (gm < M) __builtin_prefetch(Ab + (long long)gm * lda + (k0 + 16), 0, 1);
      } else if (bNT) {
        int gn = n0 + (t - 128);
        if (gn < N) __builtin_prefetch(Bb + (long long)gn * ldb + (k0 + 16), 0, 1);
      } else {
        int kk = (t - 128) >> 3, seg = (t - 128) & 7;  // 16 k-rows x 8 segments
        int gk = k0 + 16 + kk;
        if (gk < K && n0 + seg * 16 < N)
          __builtin_prefetch(Bb + (long long)gk * ldb + n0 + seg * 16, 0, 1);
      }
    }
    __syncthreads();

#pragma unroll
    for (int kk = 0; kk < 16; kk += 4) {
      int k2 = (kk + kh) >> 1;   // kk even, kh in {0,2} -> exact pair index
      const v2f* pa = (const v2f*)(As + k2 * KRS);
      const v2f* pb = (const v2f*)(Bs + k2 * KRS);
      v2f af0 = pa[wm * 32 + lr];
      v2f af1 = pa[wm * 32 + 16 + lr];
      v2f bf0 = pb[wn * 64 + lr];
      v2f bf1 = pb[wn * 64 + 16 + lr];
      v2f bf2 = pb[wn * 64 + 32 + lr];
      v2f bf3 = pb[wn * 64 + 48 + lr];
      acc[0][0] = __builtin_amdgcn_wmma_f32_16x16x4_f32(false, af0, false, bf0, (short)0, acc[0][0], false, false);
      acc[0][1] = __builtin_amdgcn_wmma_f32_16x16x4_f32(false, af0, false, bf1, (short)0, acc[0][1], false, false);
      acc[0][2] = __builtin_amdgcn_wmma_f32_16x16x4_f32(false, af0, false, bf2, (short)0, acc[0][2], false, false);
      acc[0][3] = __builtin_amdgcn_wmma_f32_16x16x4_f32(false, af0, false, bf3, (short)0, acc[0][3], false, false);
      acc[1][0] = __builtin_amdgcn_wmma_f32_16x16x4_f32(false, af1, false, bf0, (short)0, acc[1][0], false, false);
      acc[1][1] = __builtin_amdgcn_wmma_f32_16x16x4_f32(false, af1, false, bf1, (short)0, acc[1][1], false, false);
      acc[1][2] = __builtin_amdgcn_wmma_f32_16x16x4_f32(false, af1, false, bf2, (short)0, acc[1][2], false, false);
      acc[1][3] = __builtin_amdgcn_wmma_f32_16x16x4_f32(false, af1, false, bf3, (short)0, acc[1][3], false, false);
    }
    __syncthreads();
  }

  // epilogue + writeback (C layout: acc[r] -> row r (+8 upper half-wave))
  const int lrow = (lane >> 4) << 3;
  const float* rA = rowAdd ? rowAdd + (long long)bz * rowBs : nullptr;
  const float* cA = colAdd ? colAdd + (long long)bz * colBs : nullptr;
  const float* rS = resid ? resid + (long long)bz * resBs : nullptr;
  for (int i = 0; i < 2; ++i) {
    for (int s = 0; s < 4; ++s) {
      int gc = n0 + wn * 64 + s * 16 + lr;
      if (gc >= N) continue;
      for (int r = 0; r < 8; ++r) {
        int gr = m0 + wm * 32 + i * 16 + lrow + r;
        if (gr >= M) continue;
        float v = acc[i][s][r];
        if (rA) v += rA[gr];
        if (cA) v += cA[gc];
        if (rS) v += rS[(long long)gr * ldres + gc];
        if (act == 1) v = 0.5f * v * (1.0f + erff(v * 0.7071067811865476f));
        Cb[(long long)gr * ldc + gc] = v;
      }
    }
  }
}

// ---------------------------------------------------------------------------
// Rodrigues (literal, matches reference: axis = aa / max(|aa|, eps))
// ---------------------------------------------------------------------------
__device__ __forceinline__ void rodrigues_dev(float ax, float ay, float az,
                                              float* R) {
  float ang = sqrtf(ax * ax + ay * ay + az * az);
  float d = fmaxf(ang, 1e-8f);
  float x = ax / d, y = ay / d, z = az / d;
  float s = sinf(ang), c = cosf(ang), t = 1.0f - c;
  float K[9] = {0.f, -z, y, z, 0.f, -x, -y, x, 0.f};
  float K2[9];
  for (int r = 0; r < 3; ++r)
    for (int cc = 0; cc < 3; ++cc) {
      float a = 0.f;
      for (int k = 0; k < 3; ++k) a += K[r * 3 + k] * K[k * 3 + cc];
      K2[r * 3 + cc] = a;
    }
  for (int i = 0; i < 9; ++i)
    R[i] = ((i % 4 == 0) ? 1.0f : 0.0f) + s * K[i] + t * K2[i];
}

__global__ void rodrigues_kernel(const float* __restrict__ aa,
                                 float* __restrict__ rot, int n) {
  int i = blockIdx.x * blockDim.x + threadIdx.x;
  if (i >= n) return;
  rodrigues_dev(aa[i * 3], aa[i * 3 + 1], aa[i * 3 + 2], rot + (size_t)i * 9);
}

// rotate + translate points in place; layout (b,n,h,p,3), one thread per point
__global__ void apply_rigid_kernel(float* __restrict__ pts,
                                   const float* __restrict__ rot,
                                   const float* __restrict__ trans, int P,
                                   int total) {
  int i = blockIdx.x * blockDim.x + threadIdx.x;
  if (i >= total) return;
  int bn = i / (H_ * P);
  float* p = pts + (long long)i * 3;
  const float* R = rot + (long long)bn * 9;
  const float* T = trans + (long long)bn * 3;
  float x = p[0], y = p[1], z = p[2];
  p[0] = R[0] * x + R[1] * y + R[2] * z + T[0];
  p[1] = R[3] * x + R[4] * y + R[5] * z + T[1];
  p[2] = R[6] * x + R[7] * y + R[8] * z + T[2];
}

// sq_q/sq_k + per-row / per-col logit adds. i indexes (b,h,n).
__global__ void sqpts_kernel(const float* __restrict__ qp,
                             const float* __restrict__ kvp,
                             const float* __restrict__ head_w,
                             const float* __restrict__ b_b,
                             float* __restrict__ rowAdd,
                             float* __restrict__ colAdd) {
  int i = blockIdx.x * blockDim.x + threadIdx.x;
  if (i >= B_ * H_ * N_) return;
  int n = i % N_;
  int bh = i / N_;
  int h = bh % H_, b = bh / H_;
  const float* q = qp + ((long long)(b * N_ + n) * H_ + h) * (PQ_ * 3);
  float sq = 0.f;
  for (int e = 0; e < PQ_ * 3; ++e) sq += q[e] * q[e];
  const float* k = kvp + ((long long)(b * N_ + n) * H_ + h) * ((PQ_ + PV_) * 3);
  float sk = 0.f;
  for (int e = 0; e < PQ_ * 3; ++e) sk += k[e] * k[e];
  float hw = softplusf(head_w[h]) * HW_SCALE_;
  rowAdd[(long long)bh * N_ + n] = -0.5f * hw * sq + C2_ * b_b[h] - 1e9f;
  colAdd[(long long)bh * N_ + n] = -0.5f * hw * sk;
}

// Build Qcat / Kcat rows. blockIdx.x = bh*N + i  (matches GEMM batch layout).
__global__ void build_cat_kernel(const float* __restrict__ q,
                                 const float* __restrict__ kv,
                                 const float* __restrict__ qp,
                                 const float* __restrict__ kvp,
                                 const float* __restrict__ z1,
                                 const float* __restrict__ z2,
                                 const float* __restrict__ w_b,
                                 const float* __restrict__ head_w,
                                 const float* __restrict__ mask,
                                 float* __restrict__ Qc,
                                 float* __restrict__ Kc) {
  int row = blockIdx.x;
  int i = row % N_;
  int bh = row / N_;
  int h = bh % H_, b = bh / H_;
  long long ro = (long long)row * KC_;
  long long bnh = (long long)(b * N_ + i) * H_ + h;
  float hw = softplusf(head_w[h]) * HW_SCALE_;
  for (int kc = threadIdx.x; kc < KC_; kc += 128) {
    float qv, kvv;
    if (kc < 128) {
      qv = C1_ * q[bnh * C_ + kc];
      kvv = kv[bnh * (2 * C_) + kc];
    } else if (kc < 152) {
      qv = hw * qp[bnh * (PQ_ * 3) + (kc - 128)];
      kvv = kvp[bnh * ((PQ_ + PV_) * 3) + (kc - 128)];
    } else if (kc < 408) {
      int rc = kc - 152;
      qv = C2_ * z1[(long long)(b * N_ + i) * 256 + rc];
      kvv = z2[(long long)(b * N_ + i) * 256 + rc] * w_b[(rc & 127) * H_ + h];
    } else {
      float mv = mask[b * N_ + i];
      qv = 1e9f * mv;
      kvv = mv;
    }
    Qc[ro + kc] = qv;
    Kc[ro + kc] = kvv;
  }
}

// VcatT[bh][cidx][j]  (j contiguous -> NT operand of apply GEMM)
__global__ void build_vcatT_kernel(const float* __restrict__ kv,
                                   const float* __restrict__ kvp,
                                   const float* __restrict__ z2,
                                   float* __restrict__ Vc) {
  int j = blockIdx.x * 128 + threadIdx.x;
  int cidx = blockIdx.y;
  int bh = blockIdx.z;
  int h = bh % H_, b = bh / H_;
  long long bnh = (long long)(b * N_ + j) * H_ + h;
  float v;
  if (cidx < 128)
    v = kv[bnh * (2 * C_) + C_ + cidx];
  else if (cidx < 164)
    v = kvp[bnh * ((PQ_ + PV_) * 3) + PQ_ * 3 + (cidx - 128)];
  else
    v = z2[(long long)(b * N_ + j) * 256 + (cidx - 164)];
  Vc[((long long)bh * VC_ + cidx) * N_ + j] = v;
}

__global__ void softmax_kernel(float* __restrict__ a, int n) {
  __shared__ float red[256];
  float* p = a + (long long)blockIdx.x * n;
  int t = threadIdx.x;
  float mx = -1e30f;
  for (int j = t; j < n; j += 256) mx = fmaxf(mx, p[j]);
  red[t] = mx;
  __syncthreads();
  for (int s = 128; s > 0; s >>= 1) {
    if (t < s) red[t] = fmaxf(red[t], red[t + s]);
    __syncthreads();
  }
  mx = red[0];
  __syncthreads();
  float sm = 0.f;
  for (int j = t; j < n; j += 256) {
    float e = expf(p[j] - mx);
    p[j] = e;
    sm += e;
  }
  red[t] = sm;
  __syncthreads();
  for (int s = 128; s > 0; s >>= 1) {
    if (t < s) red[t] += red[t + s];
    __syncthreads();
  }
  float inv = 1.0f / red[0];
  for (int j = t; j < n; j += 256) p[j] *= inv;
}

// Assemble feats row per (b,i) from Ocat (+ invert frame on o_pt, norms, o_pair)
__global__ void feats_kernel(const float* __restrict__ Oc,
                             const float* __restrict__ z1,
                             const float* __restrict__ rot,
                             const float* __restrict__ trans,
                             float* __restrict__ feats) {
  int bi = blockIdx.x;
  int b = bi / N_, i = bi % N_;
  int t = threadIdx.x;
  long long fr = (long long)bi * 2432;
  const float* R = rot + (long long)bi * 9;
  const float* T = trans + (long long)bi * 3;
  for (int idx = t; idx < 1024; idx += 128) {
    int h = idx >> 7, c = idx & 127;
    feats[fr + idx] = Oc[((long long)(b * H_ + h) * N_ + i) * VC_ + c];
  }
  for (int idx = t; idx < 96; idx += 128) {
    int h = idx / PV_, p = idx % PV_;
    long long base = ((long long)(b * H_ + h) * N_ + i) * VC_ + 128 + p * 3;
    float dx = Oc[base] - T[0], dy = Oc[base + 1] - T[1], dz = Oc[base + 2] - T[2];
    float lx = R[0] * dx + R[3] * dy + R[6] * dz;   // rot^T * d
    float ly = R[1] * dx + R[4] * dy + R[7] * dz;
    float lz = R[2] * dx + R[5] * dy + R[8] * dz;
    feats[fr + 1024 + idx] = lx;
    feats[fr + 1120 + idx] = ly;
    feats[fr + 1216 + idx] = lz;
    feats[fr + 1312 + idx] = sqrtf(lx * lx + ly * ly + lz * lz + 1e-8f);
  }
  for (int idx = t; idx < 1024; idx += 128) {
    int h = idx >> 7, c = idx & 127;
    long long base = ((long long)(b * H_ + h) * N_ + i) * VC_ + 164;
    long long zb = (long long)bi * 256;
    feats[fr + 1408 + idx] =
        Oc[base + c] * z1[zb + c] + Oc[base + 128 + c] * z1[zb + 128 + c];
  }
}

__global__ void ln_kernel(const float* __restrict__ x,
                          const float* __restrict__ g,
                          const float* __restrict__ bb,
                          float* __restrict__ y, int D) {
  __shared__ float red[128];
  int row = blockIdx.x, t = threadIdx.x;
  const float* xr = x + (long long)row * D;
  float s = 0.f;
  for (int j = t; j < D; j += 128) s += xr[j];
  red[t] = s;
  __syncthreads();
  for (int st = 64; st > 0; st >>= 1) {
    if (t < st) red[t] += red[t + st];
    __syncthreads();
  }
  float m = red[0] / D;
  __syncthreads();
  float v = 0.f;
  for (int j = t; j < D; j += 128) {
    float d = xr[j] - m;
    v += d * d;
  }
  red[t] = v;
  __syncthreads();
  for (int st = 64; st > 0; st >>= 1) {
    if (t < st) red[t] += red[t + st];
    __syncthreads();
  }
  float inv = rsqrtf(red[0] / D + 1e-5f);
  for (int j = t; j < D; j += 128)
    y[(long long)row * D + j] = (xr[j] - m) * inv * g[j] + bb[j];
}

__global__ void backbone_kernel(const float* __restrict__ upd,
                                const float* __restrict__ rot,
                                const float* __restrict__ trans,
                                float* __restrict__ rot_out,
                                float* __restrict__ tr_out, int total) {
  int i = blockIdx.x * blockDim.x + threadIdx.x;
  if (i >= total) return;
  const float* u = upd + (long long)i * 6;
  float ax = u[0], ay = u[1], az = u[2];
  float tx = u[3], ty = u[4], tz = u[5];
  float ang = sqrtf(ax * ax + ay * ay + az * az);
  float f = fminf(ang / MAX_ANG_, 1.0f);
  ax *= f; ay *= f; az *= f;
  float tn = sqrtf(tx * tx + ty * ty + tz * tz);
  float g = fminf(tn / MAX_TR_, 1.0f);
  tx *= g; ty *= g; tz *= g;
  float dR[9];
  rodrigues_dev(ax, ay, az, dR);
  const float* R = rot + (long long)i * 9;
  float* Ro = rot_out + (long long)i * 9;
  for (int r = 0; r < 3; ++r)
    for (int c = 0; c < 3; ++c) {
      float a = 0.f;
      for (int k = 0; k < 3; ++k) a += R[r * 3 + k] * dR[k * 3 + c];
      Ro[r * 3 + c] = a;
    }
  const float* T = trans + (long long)i * 3;
  float* To = tr_out + (long long)i * 3;
  To[0] = R[0] * tx + R[1] * ty + R[2] * tz + T[0];
  To[1] = R[3] * tx + R[4] * ty + R[5] * tz + T[1];
  To[2] = R[6] * tx + R[7] * ty + R[8] * tz + T[2];
}

// ---------------------------------------------------------------------------
static inline void launch_gemm(const float* A, long long Abs, int lda,
                               const float* Bm, long long Bbs, int ldb, int bNT,
                               float* Cm, long long Cbs, int ldc, int M, int N,
                               int K, int batches, const float* colA,
                               long long colBs, const float* rowA,
                               long long rowBs, const float* resid,
                               long long resBs, int ldres, int act,
                               hipStream_t st) {
  dim3 g((N + 127) / 128, (M + 127) / 128, batches);
  wmma_gemm_f32<<<g, 256, 0, st>>>(A, Abs, lda, Bm, Bbs, ldb, bNT, Cm, Cbs,
                                   ldc, M, N, K, colA, colBs, rowA, rowBs,
                                   resid, resBs, ldres, act);
}

extern "C" void kernel_launch(void* const* d_in, const int* in_sizes, int n_in,
                              void* d_out, int out_size, void* d_ws,
                              size_t ws_size, hipStream_t stream) {
  const float* s     = (const float*)d_in[0];
  const float* r_aa  = (const float*)d_in[1];
  const float* r_tr  = (const float*)d_in[2];
  const float* z1    = (const float*)d_in[3];
  const float* z2    = (const float*)d_in[4];
  const float* mask  = (const float*)d_in[5];
  const float* w_q   = (const float*)d_in[6];
  const float* b_q   = (const float*)d_in[7];
  const float* w_kv  = (const float*)d_in[8];
  const float* b_kv  = (const float*)d_in[9];
  const float* w_qp  = (const float*)d_in[10];
  const float* b_qp  = (const float*)d_in[11];
  const float* w_kvp = (const float*)d_in[12];
  const float* b_kvp = (const float*)d_in[13];
  const float* w_b   = (const float*)d_in[14];
  const float* b_b   = (const float*)d_in[15];
  const float* hdw   = (const float*)d_in[16];
  const float* w_out = (const float*)d_in[17];
  const float* b_out = (const float*)d_in[18];
  const float* bulng = (const float*)d_in[19];
  const float* bulnb = (const float*)d_in[20];
  const float* bu_w1 = (const float*)d_in[21];
  const float* bu_b1 = (const float*)d_in[22];
  const float* bu_w2 = (const float*)d_in[23];
  const float* bu_b2 = (const float*)d_in[24];
  const float* flng  = (const float*)d_in[25];
  const float* flnb  = (const float*)d_in[26];
  const float* f_w1  = (const float*)d_in[27];
  const float* f_b1  = (const float*)d_in[28];
  const float* f_w2  = (const float*)d_in[29];
  const float* f_b2  = (const float*)d_in[30];

  float* W = (float*)d_ws;
  float* rot   = W + OFF_ROT;
  float* q     = W + OFF_Q;
  float* kv    = W + OFF_KV;
  float* qp    = W + OFF_QP;
  float* kvp   = W + OFF_KVP;
  float* rowA  = W + OFF_ROWA;
  float* colA  = W + OFF_COLA;
  float* Qc    = W + OFF_QC;
  float* Kc    = W + OFF_KC;
  float* attn  = W + OFF_A;
  float* Vc    = W + OFF_VC;
  float* Oc    = W + OFF_OC;
  float* feats = W + OFF_F;
  float* lnb   = W + OFF_LN;
  float* buh   = W + OFF_BH;
  float* upd   = W + OFF_UPD;
  float* ffh   = W + OFF_FH;

  float* outS  = (float*)d_out;                 // (B,N,384)
  float* outR  = outS + (size_t)B_ * N_ * CS_;  // (B,N,3,3)
  float* outT  = outR + (size_t)B_ * N_ * 9;    // (B,N,3)

  const int BN = B_ * N_;       // 2048
  const int BH = B_ * H_;       // 16
  const long long LL0 = 0;

  // 1) rigid frames
  rodrigues_kernel<<<(BN + 255) / 256, 256, 0, stream>>>(r_aa, rot, BN);

  // 2) input projections (WMMA GEMMs)
  launch_gemm(s, LL0, CS_, w_q, LL0, H_ * C_, 0, q, LL0, H_ * C_, BN, H_ * C_,
              CS_, 1, b_q, LL0, nullptr, LL0, nullptr, LL0, 0, 0, stream);
  launch_gemm(s, LL0, CS_, w_kv, LL0, 2 * H_ * C_, 0, kv, LL0, 2 * H_ * C_, BN,
              2 * H_ * C_, CS_, 1, b_kv, LL0, nullptr, LL0, nullptr, LL0, 0, 0,
              stream);
  launch_gemm(s, LL0, CS_, w_qp, LL0, H_ * PQ_ * 3, 0, qp, LL0, H_ * PQ_ * 3,
              BN, H_ * PQ_ * 3, CS_, 1, b_qp, LL0, nullptr, LL0, nullptr, LL0,
              0, 0, stream);
  launch_gemm(s, LL0, CS_, w_kvp, LL0, H_ * (PQ_ + PV_) * 3, 0, kvp, LL0,
              H_ * (PQ_ + PV_) * 3, BN, H_ * (PQ_ + PV_) * 3, CS_, 1, b_kvp,
              LL0, nullptr, LL0, nullptr, LL0, 0, 0, stream);

  // 3) rotate + translate points (in place)
  apply_rigid_kernel<<<(BN * H_ * PQ_ + 255) / 256, 256, 0, stream>>>(
      qp, rot, r_tr, PQ_, BN * H_ * PQ_);
  apply_rigid_kernel<<<(BN * H_ * (PQ_ + PV_) + 255) / 256, 256, 0, stream>>>(
      kvp, rot, r_tr, PQ_ + PV_, BN * H_ * (PQ_ + PV_));

  // 4) row/col logit adds (sq terms, head bias, mask constant)
  sqpts_kernel<<<(BH * N_ + 255) / 256, 256, 0, stream>>>(qp, kvp, hdw, b_b,
                                                          rowA, colA);

  // 5) concatenated Q/K operands (K = 409)
  build_cat_kernel<<<BH * N_, 128, 0, stream>>>(q, kv, qp, kvp, z1, z2, w_b,
                                                hdw, mask, Qc, Kc);

  // 6) attention logits: batched (b,h) GEMM  a = Qcat * Kcat^T + row + col
  launch_gemm(Qc, (long long)N_ * KC_, KC_, Kc, (long long)N_ * KC_, KC_, 1,
              attn, (long long)N_ * N_, N_, N_, N_, KC_, BH, colA,
              (long long)N_, rowA, (long long)N_, nullptr, LL0, 0, 0, stream);

  // 7) softmax over j (in place; 64 MB -> L2 resident)
  softmax_kernel<<<BH * N_, 256, 0, stream>>>(attn, N_);

  // 8) concatenated V-side operand (420 x N, K contiguous)
  {
    dim3 g(N_ / 128, VC_, BH);
    build_vcatT_kernel<<<g, 128, 0, stream>>>(kv, kvp, z2, Vc);
  }

  // 9) apply attention: Ocat = a * Vcat   (batched, K = 1024)
  launch_gemm(attn, (long long)N_ * N_, N_, Vc, (long long)VC_ * N_, N_, 1, Oc,
              (long long)N_ * VC_, VC_, N_, VC_, N_, BH, nullptr, LL0, nullptr,
              LL0, nullptr, LL0, 0, 0, stream);

  // 10) feats assembly (frame inverse on o_pt, norms, pair contraction)
  feats_kernel<<<BN, 128, 0, stream>>>(Oc, z1, rot, r_tr, feats);

  // 11) output projection + residual: s1 = s + feats @ w_out + b_out
  launch_gemm(feats, LL0, 2432, w_out, LL0, CS_, 0, outS, LL0, CS_, BN, CS_,
              2432, 1, b_out, LL0, nullptr, LL0, s, LL0, CS_, 0, stream);

  // 12) backbone update branch
  ln_kernel<<<BN, 128, 0, stream>>>(outS, bulng, bulnb, lnb, CS_);
  launch_gemm(lnb, LL0, CS_, bu_w1, LL0, 128, 0, buh, LL0, 128, BN, 128, CS_,
              1, bu_b1, LL0, nullptr, LL0, nullptr, LL0, 0, 1, stream);
  launch_gemm(buh, LL0, 128, bu_w2, LL0, 6, 0, upd, LL0, 6, BN, 6, 128, 1,
              bu_b2, LL0, nullptr, LL0, nullptr, LL0, 0, 0, stream);
  backbone_kernel<<<(BN + 255) / 256, 256, 0, stream>>>(upd, rot, r_tr, outR,
                                                        outT, BN);

  // 13) FFN branch: s = s1 + W2(gelu(W1 ln(s1)))
  ln_kernel<<<BN, 128, 0, stream>>>(outS, flng, flnb, lnb, CS_);
  launch_gemm(lnb, LL0, CS_, f_w1, LL0, 4 * CS_, 0, ffh, LL0, 4 * CS_, BN,
              4 * CS_, CS_, 1, f_b1, LL0, nullptr, LL0, nullptr, LL0, 0, 1,
              stream);
  launch_gemm(ffh, LL0, 4 * CS_, f_w2, LL0, CS_, 0, outS, LL0, CS_, BN, CS_,
              4 * CS_, 1, f_b2, LL0, nullptr, LL0, outS, LL0, CS_, 0, stream);
}